// ScannedRNN_68504728371782
// MI455X (gfx1250) — compile-verified
//
#include <hip/hip_runtime.h>
#include <stdint.h>

typedef __attribute__((ext_vector_type(2))) float v2f;
typedef __attribute__((ext_vector_type(8))) float v8f;

#define T_STEPS 1024
#define BATCH   128
#define HID     256
#define N4      1024
#define NBLOCKS 16
#define LDS_STRIDE 260  // 256 + 4 floats pad -> conflict-free ds_load_b64 A-frag reads

__device__ __forceinline__ float sigmoid_f(float x) {
  return 1.0f / (1.0f + __expf(-x));
}

// CDNA5 async global->LDS copy (ASYNCcnt-tracked, no VGPR data movement).
__device__ __forceinline__ void async_b128(uint32_t lds_addr, const float* g) {
  asm volatile("global_load_async_to_lds_b128 %0, %1, off"
               :: "v"(lds_addr), "v"(g) : "memory");
}
__device__ __forceinline__ void wait_async0() {
  asm volatile("s_wait_asynccnt 0x0" ::: "memory");
}

// Stage 16 rows x 256 floats (row stride HID in global) into an LDS tile with
// row stride LDS_STRIDE. 256 threads: 16 threads per row, 64B (4x b128) each.
__device__ __forceinline__ void stage_tile(float* sh, const float* gbase) {
  const int row = threadIdx.x >> 4;
  const int q   = threadIdx.x & 15;
  const float*  g = gbase + row * HID + q * 16;
  const uint32_t l =
      (uint32_t)(uintptr_t)(sh + row * LDS_STRIDE + q * 16);  // addr[31:0] = LDS offset
#pragma unroll
  for (int i = 0; i < 4; ++i) async_b128(l + i * 16, g + i * 4);
}

// Persistent fused LSTM scan. 16 blocks x 256 threads = 128 wave32 waves.
// Wave w owns batch rows m0..m0+15 (uniform per block) and hidden cols
// j0..j0+15 across all four gates, so gate math is lane-local and c lives in
// 8 VGPRs per lane for the whole scan. h/x tiles are staged once per WGP via
// async-LDS; x@Wi for step t+1 is computed in the grid-barrier slack.
__global__ __launch_bounds__(256, 1) void lstm_scan_kernel(
    const float* __restrict__ ins, const unsigned char* __restrict__ resets,
    const float* __restrict__ Wi, const float* __restrict__ Wh,
    const float* __restrict__ bias, const float* __restrict__ c0,
    float* __restrict__ hbuf, unsigned int* __restrict__ counter,
    float* __restrict__ ys)
{
  __shared__ __align__(16) float sh_h[16 * LDS_STRIDE];
  __shared__ __align__(16) float sh_x[16 * LDS_STRIDE];

  const int lane = threadIdx.x & 31;
  const int wv   = (blockIdx.x << 3) + (threadIdx.x >> 5);  // 0..127
  const int m0   = (wv >> 4) << 4;   // batch tile origin; uniform per block
  const int j0   = (wv & 15) << 4;   // hidden-col tile origin
  const int half = lane >> 4;
  const int lcol = lane & 15;
  const int koff = half << 1;        // A-frag K sub-offset {0,2} per ISA layout

  // Persistent cell state: element (m0 + v + 8*half, j0 + lcol) per C/D layout.
  float creg[8];
#pragma unroll
  for (int v = 0; v < 8; ++v)
    creg[v] = c0[(m0 + v + 8 * half) * HID + j0 + lcol];

  float bg[4];
#pragma unroll
  for (int g = 0; g < 4; ++g) bg[g] = bias[g * HID + j0 + lcol];

  const float* arowlds = sh_h + lcol * LDS_STRIDE;  // this lane's A row in LDS
  const float* xrowlds = sh_x + lcol * LDS_STRIDE;

  // ---- prologue: stage x(0) and compute xacc(0) = b + x(0) @ Wi ----
  stage_tile(sh_x, ins + m0 * HID);
  wait_async0();
  __syncthreads();

  v8f xacc[4];
#pragma unroll
  for (int g = 0; g < 4; ++g)
#pragma unroll
    for (int v = 0; v < 8; ++v) xacc[g][v] = bg[g];
  for (int k0 = 0; k0 < HID; k0 += 4) {
    v2f a = *(const v2f*)(xrowlds + k0 + koff);             // ds_load_b64
    const float* w0 = Wi + (size_t)(k0 + koff) * N4 + j0 + lcol;
#pragma unroll
    for (int g = 0; g < 4; ++g) {
      v2f bf; bf.x = w0[g * HID]; bf.y = w0[g * HID + N4];
      xacc[g] = __builtin_amdgcn_wmma_f32_16x16x4_f32(
          false, a, false, bf, (short)0, xacc[g], false, false);
    }
  }

  for (int t = 0; t < T_STEPS; ++t) {
    const unsigned char* rt = resets + t * BATCH;

    // ---- wait until h_t is fully published by all blocks ----
    if (t > 0 && threadIdx.x == 0) {
      const unsigned target = (unsigned)NBLOCKS * (unsigned)t;
      while (*(volatile unsigned int*)counter < target)
        __builtin_amdgcn_s_sleep(1);
      __threadfence();
    }
    __syncthreads();

    // ---- stage h_t tile once per WGP via async-LDS ----
    stage_tile(sh_h, hbuf + (t & 1) * (BATCH * HID) + m0 * HID);
    wait_async0();
    __syncthreads();

    const float amask = rt[m0 + lcol] ? 0.0f : 1.0f;  // reset masks h before z

    v8f acc[4];
#pragma unroll
    for (int g = 0; g < 4; ++g) acc[g] = xacc[g];

    // ---- z += (reset-masked h) @ Wh : 64 k-steps x 4 gate WMMAs ----
    for (int k0 = 0; k0 < HID; k0 += 4) {
      v2f a = *(const v2f*)(arowlds + k0 + koff);           // ds_load_b64
      a.x *= amask; a.y *= amask;
      const float* w0 = Wh + (size_t)(k0 + koff) * N4 + j0 + lcol;
#pragma unroll
      for (int g = 0; g < 4; ++g) {
        v2f bf; bf.x = w0[g * HID]; bf.y = w0[g * HID + N4];
        acc[g] = __builtin_amdgcn_wmma_f32_16x16x4_f32(
            false, a, false, bf, (short)0, acc[g], false, false);
      }
    }

    // ---- gates + state update (all 4 gates share (row,col) per lane) ----
    float* hnxt = hbuf + ((t + 1) & 1) * (BATCH * HID);
#pragma unroll
    for (int v = 0; v < 8; ++v) {
      const int row = m0 + v + 8 * half;
      const float cm = rt[row] ? 0.0f : 1.0f;
      const float iv = acc[0][v], fv = acc[1][v];
      const float gv = acc[2][v], ov = acc[3][v];
      const float nc = sigmoid_f(fv) * (creg[v] * cm) + sigmoid_f(iv) * tanhf(gv);
      const float nh = sigmoid_f(ov) * tanhf(nc);
      creg[v] = nc;
      hnxt[row * HID + j0 + lcol] = nh;
      ys[((size_t)t * BATCH + row) * HID + j0 + lcol] = nh;
    }
    __threadfence();     // publish this thread's h_{t+1} stores to device scope
    __syncthreads();     // whole block done writing h_{t+1} / reading sh_h
    if (threadIdx.x == 0) atomicAdd(counter, 1u);  // arrive for step t+1

    // ---- overlap barrier slack: xacc(t+1) = b + x(t+1) @ Wi ----
    if (t + 1 < T_STEPS) {
      stage_tile(sh_x, ins + (size_t)(t + 1) * (BATCH * HID) + m0 * HID);
      wait_async0();
      __syncthreads();
#pragma unroll
      for (int g = 0; g < 4; ++g)
#pragma unroll
        for (int v = 0; v < 8; ++v) xacc[g][v] = bg[g];
      for (int k0 = 0; k0 < HID; k0 += 4) {
        v2f a = *(const v2f*)(xrowlds + k0 + koff);
        const float* w0 = Wi + (size_t)(k0 + koff) * N4 + j0 + lcol;
#pragma unroll
        for (int g = 0; g < 4; ++g) {
          v2f bf; bf.x = w0[g * HID]; bf.y = w0[g * HID + N4];
          xacc[g] = __builtin_amdgcn_wmma_f32_16x16x4_f32(
              false, a, false, bf, (short)0, xacc[g], false, false);
        }
      }
    }
  }
}

extern "C" void kernel_launch(void* const* d_in, const int* in_sizes, int n_in,
                              void* d_out, int out_size, void* d_ws, size_t ws_size,
                              hipStream_t stream) {
  const float*         ins    = (const float*)d_in[0];
  const unsigned char* resets = (const unsigned char*)d_in[1];
  const float*         Wi     = (const float*)d_in[2];
  const float*         Wh     = (const float*)d_in[3];
  const float*         bias   = (const float*)d_in[4];
  const float*         c0     = (const float*)d_in[5];
  const float*         h0     = (const float*)d_in[6];
  float* ys = (float*)d_out;

  // ws layout: [ h double buffer: 2*B*H floats ][ barrier counter ]
  float* hbuf = (float*)d_ws;
  unsigned int* counter =
      (unsigned int*)((char*)d_ws + (size_t)2 * BATCH * HID * sizeof(float));

  hipMemsetAsync(counter, 0, sizeof(unsigned int), stream);
  hipMemcpyAsync(hbuf, h0, (size_t)BATCH * HID * sizeof(float),
                 hipMemcpyDeviceToDevice, stream);

  lstm_scan_kernel<<<NBLOCKS, 256, 0, stream>>>(
      ins, resets, Wi, Wh, bias, c0, hbuf, counter, ys);

  (void)in_sizes; (void)n_in; (void)out_size; (void)ws_size;
}